// GAT_47210280518300
// MI455X (gfx1250) — compile-verified
//
#include <hip/hip_runtime.h>
#include <hip/hip_bf16.h>
#include <math.h>

typedef float v2f __attribute__((ext_vector_type(2)));
typedef float v8f __attribute__((ext_vector_type(8)));

#define NEG_SLOPE 0.2f

// ---------------------------------------------------------------------------
// Utility
// ---------------------------------------------------------------------------
__device__ __forceinline__ void atomicMaxF(float* addr, float val) {
    // Classic order-preserving bit trick: signed-int max for >=0,
    // unsigned-int min for <0. Works with -inf initialization.
    if (val >= 0.0f) {
        atomicMax((int*)addr, __float_as_int(val));
    } else {
        atomicMin((unsigned int*)addr, __float_as_uint(val));
    }
}

__device__ __forceinline__ void edge_sd(const int* __restrict__ ei, int e, int E,
                                        int& s, int& d) {
    if (e < E) { s = ei[e]; d = ei[E + e]; }  // edge_index[0][e], edge_index[1][e]
    else       { s = d = e - E; }             // self loop
}

__global__ void fill_kernel(float* __restrict__ p, float v, int n) {
    int i = blockIdx.x * blockDim.x + threadIdx.x;
    if (i < n) p[i] = v;
}

// ---------------------------------------------------------------------------
// FP32 WMMA GEMM:  H[N x Fout] = X[N x K] @ W[K x Fout]
// One wave (32 lanes) per 16x16 output tile; K consumed 4 at a time via
// v_wmma_f32_16x16x4_f32.  Requires N % 16 == 0, Fout = 16*NCOLT, K % 4 == 0.
// A tile staged in LDS (padded to kill bank conflicts).
// ---------------------------------------------------------------------------
template <int K, int NCOLT>
__global__ void gemm_wmma_f32(const float* __restrict__ X,
                              const float* __restrict__ W,
                              float* __restrict__ H) {
    constexpr int FOUT = 16 * NCOLT;
    __shared__ float As[16][K + 4];

    const int r0   = blockIdx.x * 16;
    const int lane = threadIdx.x;          // 0..31
    const int wave = threadIdx.y;          // 0..NCOLT-1 -> column tile
    const int tid  = wave * 32 + lane;
    const int nthr = 32 * NCOLT;

    // Cooperative, coalesced load of the 16 x K A-tile into LDS.
    for (int i = tid; i < 16 * K; i += nthr) {
        int rr = i / K, cc = i % K;
        As[rr][cc] = X[(size_t)(r0 + rr) * K + cc];
    }
    __syncthreads();

    const int half = lane >> 4;            // 0: lanes 0-15, 1: lanes 16-31
    const int m    = lane & 15;            // row (A) / col (B,C,D) index
    const int c0   = wave * 16;

    v8f acc = {};
    #pragma unroll 4
    for (int kk = 0; kk < K; kk += 4) {
        const int kb = kk + 2 * half;
        v2f a, b;
        // A (16x4): lanes 0-15 hold K=kk..kk+1, lanes 16-31 hold K=kk+2..kk+3
        a[0] = As[m][kb];
        a[1] = As[m][kb + 1];
        // B (4x16): same K split across lane halves, column = lane%16
        b[0] = W[(size_t)(kb)     * FOUT + c0 + m];
        b[1] = W[(size_t)(kb + 1) * FOUT + c0 + m];
        acc = __builtin_amdgcn_wmma_f32_16x16x4_f32(
            /*neg_a=*/false, a, /*neg_b=*/false, b,
            /*c_mod=*/(short)0, acc, /*reuse_a=*/false, /*reuse_b=*/false);
    }

    // C/D layout: VGPR r -> row (r + 8*half), col = lane%16
    #pragma unroll
    for (int r = 0; r < 8; ++r) {
        H[(size_t)(r0 + r + 8 * half) * FOUT + c0 + m] = acc[r];
    }
}

// ---------------------------------------------------------------------------
// Per-node attention logits: as[n,h] = sum_d h[n,h,d]*a_src[h,d]  (same for ad)
// ---------------------------------------------------------------------------
template <int H, int D>
__global__ void alpha_kernel(const float* __restrict__ h,
                             const float* __restrict__ a_src,
                             const float* __restrict__ a_dst,
                             float* __restrict__ as, float* __restrict__ ad,
                             int n) {
    int i = blockIdx.x * blockDim.x + threadIdx.x;   // over n*H
    if (i >= n * H) return;
    int node = i / H, hh = i % H;
    const float* hp = h + (size_t)node * (H * D) + hh * D;
    float s = 0.f, d = 0.f;
    #pragma unroll
    for (int j = 0; j < D; ++j) {
        float v = hp[j];
        s += v * a_src[hh * D + j];
        d += v * a_dst[hh * D + j];
    }
    as[i] = s;
    ad[i] = d;
}

// ---------------------------------------------------------------------------
// Edge pass A: e = leaky_relu(as[src]+ad[dst]); store e; segment max into m.
// ---------------------------------------------------------------------------
template <int H>
__global__ void edge_max_kernel(const int* __restrict__ ei, int E, int N,
                                const float* __restrict__ as,
                                const float* __restrict__ ad,
                                float* __restrict__ ebuf,
                                float* __restrict__ m) {
    int i = blockIdx.x * blockDim.x + threadIdx.x;   // over (E+N)*H
    if (i >= (E + N) * H) return;
    int e = i / H, hh = i % H;
    int s, d;
    edge_sd(ei, e, E, s, d);
    float v = as[s * H + hh] + ad[d * H + hh];
    v = (v > 0.f) ? v : NEG_SLOPE * v;
    ebuf[i] = v;
    atomicMaxF(&m[d * H + hh], v);
}

// ---------------------------------------------------------------------------
// Edge pass B: ex = exp(e - m[dst]); store ex; segment sum into den.
// ---------------------------------------------------------------------------
template <int H>
__global__ void edge_exp_kernel(const int* __restrict__ ei, int E, int N,
                                float* __restrict__ ebuf,
                                const float* __restrict__ m,
                                float* __restrict__ den) {
    int i = blockIdx.x * blockDim.x + threadIdx.x;
    if (i >= (E + N) * H) return;
    int e = i / H, hh = i % H;
    int s, d;
    edge_sd(ei, e, E, s, d);
    float ex = expf(ebuf[i] - m[d * H + hh]);
    ebuf[i] = ex;
    atomicAdd(&den[d * H + hh], ex);
}

// ---------------------------------------------------------------------------
// Edge pass C: alpha = ex/(den[dst]+1e-16); agg[dst] += h[src]*alpha
// One thread per (edge, head), 16 atomics each.
// ---------------------------------------------------------------------------
template <int H, int D>
__global__ void edge_msg_kernel(const int* __restrict__ ei, int E, int N,
                                const float* __restrict__ h,
                                const float* __restrict__ ebuf,
                                const float* __restrict__ den,
                                float* __restrict__ agg) {
    int i = blockIdx.x * blockDim.x + threadIdx.x;
    if (i >= (E + N) * H) return;
    int e = i / H, hh = i % H;
    int s, d;
    edge_sd(ei, e, E, s, d);
    float alpha = ebuf[i] / (den[d * H + hh] + 1e-16f);
    const float* hp = h + (size_t)s * (H * D) + hh * D;
    float* ap = agg + (size_t)d * (H * D) + hh * D;
    #pragma unroll
    for (int j = 0; j < D; ++j) {
        atomicAdd(&ap[j], hp[j] * alpha);
    }
}

// ---------------------------------------------------------------------------
// out = elu(agg + b)  (in place, layer-1 output)
// ---------------------------------------------------------------------------
__global__ void bias_elu_kernel(float* __restrict__ agg,
                                const float* __restrict__ b, int n, int F) {
    int i = blockIdx.x * blockDim.x + threadIdx.x;
    if (i >= n * F) return;
    float v = agg[i] + b[i % F];
    agg[i] = (v > 0.f) ? v : (expf(v) - 1.f);
}

// ---------------------------------------------------------------------------
// out += b  (final bias)
// ---------------------------------------------------------------------------
__global__ void add_bias_kernel(float* __restrict__ out,
                                const float* __restrict__ b, int n, int F) {
    int i = blockIdx.x * blockDim.x + threadIdx.x;
    if (i >= n * F) return;
    out[i] += b[i % F];
}

// ---------------------------------------------------------------------------
// Host orchestration
// ---------------------------------------------------------------------------
static inline int cdiv(int a, int b) { return (a + b - 1) / b; }

extern "C" void kernel_launch(void* const* d_in, const int* in_sizes, int n_in,
                              void* d_out, int out_size, void* d_ws, size_t ws_size,
                              hipStream_t stream) {
    const float* x      = (const float*)d_in[0];
    const int*   ei     = (const int*)  d_in[1];
    const float* W1     = (const float*)d_in[2];
    const float* a_src1 = (const float*)d_in[3];
    const float* a_dst1 = (const float*)d_in[4];
    const float* b1     = (const float*)d_in[5];
    const float* W2     = (const float*)d_in[6];
    const float* a_src2 = (const float*)d_in[7];
    const float* a_dst2 = (const float*)d_in[8];
    const float* b2     = (const float*)d_in[9];

    const int N  = in_sizes[0] / 128;   // 100000
    const int E  = in_sizes[1] / 2;     // 1000000
    const int EP = E + N;               // with self loops

    // Workspace layout (floats)
    float* ws   = (float*)d_ws;
    float* h1   = ws;                    // N*64
    float* agg1 = h1   + (size_t)N * 64; // N*64  (becomes elu'd layer-1 output)
    float* as1  = agg1 + (size_t)N * 64; // N*4
    float* ad1  = as1  + (size_t)N * 4;  // N*4
    float* m1   = ad1  + (size_t)N * 4;  // N*4
    float* den1 = m1   + (size_t)N * 4;  // N*4
    float* ex1  = den1 + (size_t)N * 4;  // EP*4
    // Layer-2 scratch reuses the h1 region (free after layer-1 message pass)
    float* h2   = h1;                    // N*16
    float* as2  = h2   + (size_t)N * 16; // N
    float* ad2  = as2  + N;              // N
    float* m2   = ad2  + N;              // N
    float* den2 = m2   + N;              // N
    float* ex2  = den2 + N;              // EP

    float* out  = (float*)d_out;         // N*16

    const int TB = 256;

    // ---------------- Layer 1 ----------------
    fill_kernel<<<cdiv(N * 4, TB), TB, 0, stream>>>(m1, -INFINITY, N * 4);
    fill_kernel<<<cdiv(N * 4, TB), TB, 0, stream>>>(den1, 0.f, N * 4);
    fill_kernel<<<cdiv(N * 64, TB), TB, 0, stream>>>(agg1, 0.f, N * 64);

    // h1 = x @ W1   (K=128, 4 column tiles of 16 -> Fout=64)
    gemm_wmma_f32<128, 4><<<N / 16, dim3(32, 4), 0, stream>>>(x, W1, h1);

    alpha_kernel<4, 16><<<cdiv(N * 4, TB), TB, 0, stream>>>(
        h1, a_src1, a_dst1, as1, ad1, N);

    edge_max_kernel<4><<<cdiv(EP * 4, TB), TB, 0, stream>>>(
        ei, E, N, as1, ad1, ex1, m1);
    edge_exp_kernel<4><<<cdiv(EP * 4, TB), TB, 0, stream>>>(
        ei, E, N, ex1, m1, den1);
    edge_msg_kernel<4, 16><<<cdiv(EP * 4, TB), TB, 0, stream>>>(
        ei, E, N, h1, ex1, den1, agg1);

    bias_elu_kernel<<<cdiv(N * 64, TB), TB, 0, stream>>>(agg1, b1, N, 64);

    // ---------------- Layer 2 ----------------
    fill_kernel<<<cdiv(N, TB), TB, 0, stream>>>(m2, -INFINITY, N);
    fill_kernel<<<cdiv(N, TB), TB, 0, stream>>>(den2, 0.f, N);
    fill_kernel<<<cdiv(N * 16, TB), TB, 0, stream>>>(out, 0.f, N * 16);

    // h2 = elu_out @ W2   (K=64, 1 column tile -> Fout=16)
    gemm_wmma_f32<64, 1><<<N / 16, dim3(32, 1), 0, stream>>>(agg1, W2, h2);

    alpha_kernel<1, 16><<<cdiv(N, TB), TB, 0, stream>>>(
        h2, a_src2, a_dst2, as2, ad2, N);

    edge_max_kernel<1><<<cdiv(EP, TB), TB, 0, stream>>>(
        ei, E, N, as2, ad2, ex2, m2);
    edge_exp_kernel<1><<<cdiv(EP, TB), TB, 0, stream>>>(
        ei, E, N, ex2, m2, den2);
    edge_msg_kernel<1, 16><<<cdiv(EP, TB), TB, 0, stream>>>(
        ei, E, N, h2, ex2, den2, out);

    add_bias_kernel<<<cdiv(N * 16, TB), TB, 0, stream>>>(out, b2, N, 16);
}